// NAM_58411555226364
// MI455X (gfx1250) — compile-verified
//
#include <hip/hip_runtime.h>

typedef __attribute__((ext_vector_type(2))) float v2f;
typedef __attribute__((ext_vector_type(8))) float v8f;

#define F 256
#define H 64
#define BATCH 16384
#define NHID 3                          // hidden-to-hidden layers (DEPTH-2)
#define ROWS_PER_WAVE 32                // M per wave (two 16-row A fragments)
#define WAVES 4
#define THREADS (WAVES * 32)            // 128
#define ROWS_PER_BLOCK (ROWS_PER_WAVE * WAVES)   // 128
#define HPAD 68                         // padded row stride (floats) for LDS h tiles
#define PRSTRIDE 160                    // pair-row stride: 128 data + 32 pad (bank split)
#define NSLICES 4
#define F_PER_SLICE (F / NSLICES)

__global__ __launch_bounds__(THREADS)
void nam_wmma_kernel(const float* __restrict__ x,    // [B, F]
                     const float* __restrict__ W1,   // [F, H]
                     const float* __restrict__ b1,   // [F, H]
                     const float* __restrict__ Wh,   // [NHID, F, H, H] (h_in, h_out)
                     const float* __restrict__ bh,   // [NHID, F, H]
                     const float* __restrict__ Wo,   // [F, H]
                     const float* __restrict__ bo,   // [F]
                     float* __restrict__ dst,        // [numSlices, B] partials (or [B])
                     int fPerSlice)
{
    // Weight tile staged pair-interleaved: pr = k/2 rows of 64 (lo,hi) float pairs.
    __shared__ float sWp[32 * PRSTRIDE];            // 20 KB
    __shared__ float sW1[H];
    __shared__ float sb1[H];
    __shared__ float sbh[NHID][H];
    __shared__ float sWo[H];
    __shared__ float sbo;
    __shared__ float sx[ROWS_PER_BLOCK];
    __shared__ float sH[WAVES][ROWS_PER_WAVE * HPAD];   // 34.8 KB

    const int t    = threadIdx.x;
    const int wave = t >> 5;
    const int lane = t & 31;
    const int m    = lane & 15;          // N col in C-layout / M row in A-layout
    const int hi   = lane >> 4;          // half-wave select
    const int tileBase = blockIdx.x * ROWS_PER_BLOCK;

    const int fStart = blockIdx.y * fPerSlice;
    const int fEnd   = fStart + fPerSlice;
    float* outSlice  = dst + (size_t)blockIdx.y * BATCH;

    // p[g][r]: partial feature-sum for row (g*16 + 8*hi + r), cols {16*nb + m}.
    float p[2][8];
#pragma unroll
    for (int g = 0; g < 2; ++g)
#pragma unroll
        for (int r = 0; r < 8; ++r) p[g][r] = 0.f;
    float bo_total = 0.f;

    for (int f = fStart; f < fEnd; ++f) {
        // ---------------- per-feature small params + x column ----------------
        __syncthreads();
        if (t < H) {
            sW1[t] = W1[f * H + t];
            sb1[t] = b1[f * H + t];
            sWo[t] = Wo[f * H + t];
#pragma unroll
            for (int l = 0; l < NHID; ++l)
                sbh[l][t] = bh[((size_t)l * F + f) * H + t];
        }
        if (t == 0) sbo = bo[f];
        sx[t] = x[(size_t)(tileBase + t) * F + f];
        if (f + 1 < fEnd)   // pull next x column toward L2
            __builtin_prefetch(x + (size_t)(tileBase + t) * F + f + 1, 0, 1);
        __syncthreads();

        // ------- layer 1: h = relu(x*W1 + b1) into sH (A-layout, rows m and m+16) -------
        {
#pragma unroll
            for (int g = 0; g < 2; ++g) {
                const int row = m + 16 * g;
                const float xv = sx[wave * ROWS_PER_WAVE + row];
                float* hbase = &sH[wave][row * HPAD];
#pragma unroll
                for (int jj = 0; jj < 8; ++jj) {
                    const int k = 32 * hi + 4 * jj;
                    float4 v;
                    v.x = fmaxf(fmaf(xv, sW1[k + 0], sb1[k + 0]), 0.f);
                    v.y = fmaxf(fmaf(xv, sW1[k + 1], sb1[k + 1]), 0.f);
                    v.z = fmaxf(fmaf(xv, sW1[k + 2], sb1[k + 2]), 0.f);
                    v.w = fmaxf(fmaf(xv, sW1[k + 3], sb1[k + 3]), 0.f);
                    *(float4*)(hbase + k) = v;
                }
            }
        }

        // ---------------- hidden layers via fp32 WMMA ----------------
        v8f acc[2][4];
        for (int l = 0; l < NHID; ++l) {
            __syncthreads();   // prev sWp consumers done; sH writes visible cross-lane
            {
                // Stage Wh[l][f] pair-interleaved: (k even, k odd) value pairs per col.
                const float* src = Wh + ((size_t)l * F + f) * (H * H);
#pragma unroll
                for (int it = 0; it < 512 / THREADS; ++it) {
                    const int j  = t + it * THREADS;   // 0..511
                    const int pr = j >> 4;             // 0..31 (= k/2)
                    const int n  = (j & 15) * 4;       // col group
                    const float4 a = *(const float4*)(src + (2 * pr) * H + n);
                    const float4 b = *(const float4*)(src + (2 * pr + 1) * H + n);
                    float4 v0, v1;
                    v0.x = a.x; v0.y = b.x; v0.z = a.y; v0.w = b.y;
                    v1.x = a.z; v1.y = b.z; v1.z = a.w; v1.w = b.w;
                    float* d = &sWp[pr * PRSTRIDE + 2 * n];
                    *(float4*)(d)     = v0;
                    *(float4*)(d + 4) = v1;
                }
            }
            if (f + 1 < fEnd)   // pull next feature's weight tile toward L2
                __builtin_prefetch(Wh + ((size_t)l * F + (f + 1)) * (H * H) + t * 32, 0, 1);
            __syncthreads();

            // init accumulators with bias (C[m][n] = bh[n] for every row)
#pragma unroll
            for (int nb = 0; nb < 4; ++nb) {
                const float c = sbh[l][nb * 16 + m];
                v8f a;
#pragma unroll
                for (int r = 0; r < 8; ++r) a[r] = c;
                acc[0][nb] = a;
                acc[1][nb] = a;
            }

            const float* hbase0 = &sH[wave][m * HPAD + 2 * hi];          // rows 0..15
            const float* hbase1 = hbase0 + 16 * HPAD;                    // rows 16..31

#pragma unroll
            for (int kb = 0; kb < 16; ++kb) {
                v2f A0, A1;
                __builtin_memcpy(&A0, hbase0 + 4 * kb, sizeof(A0));      // ds_load_b64
                __builtin_memcpy(&A1, hbase1 + 4 * kb, sizeof(A1));
                const float* bb = &sWp[(2 * kb + hi) * PRSTRIDE + 2 * m];
#pragma unroll
                for (int nb = 0; nb < 4; ++nb) {
                    v2f Bv;
                    __builtin_memcpy(&Bv, bb + 32 * nb, sizeof(Bv));     // ds_load_b64, pair-ready
                    acc[0][nb] = __builtin_amdgcn_wmma_f32_16x16x4_f32(
                        false, A0, false, Bv, (short)0, acc[0][nb], false, false);
                    acc[1][nb] = __builtin_amdgcn_wmma_f32_16x16x4_f32(
                        false, A1, false, Bv, (short)0, acc[1][nb], false, false);
                }
            }

            __syncthreads();   // all sH / sWp reads complete before overwrite
            if (l < NHID - 1) {
#pragma unroll
                for (int g = 0; g < 2; ++g)
#pragma unroll
                    for (int r = 0; r < 8; ++r) {
                        float* row = &sH[wave][(16 * g + 8 * hi + r) * HPAD + m];
#pragma unroll
                        for (int nb = 0; nb < 4; ++nb)
                            row[nb * 16] = fmaxf(acc[g][nb][r], 0.f);
                    }
            }
        }

        // ---------------- output layer: dot(relu(h), Wo) ----------------
#pragma unroll
        for (int nb = 0; nb < 4; ++nb) {
            const float w = sWo[nb * 16 + m];
#pragma unroll
            for (int g = 0; g < 2; ++g)
#pragma unroll
                for (int r = 0; r < 8; ++r)
                    p[g][r] = fmaf(fmaxf(acc[g][nb][r], 0.f), w, p[g][r]);
        }
        bo_total += sbo;
    }

    // -------- reduce row partials across the 16 lanes of each half-wave --------
#pragma unroll
    for (int g = 0; g < 2; ++g)
#pragma unroll
        for (int r = 0; r < 8; ++r) {
            float v = p[g][r];
#pragma unroll
            for (int off = 1; off < 16; off <<= 1)
                v += __shfl_xor(v, off, 32);
            p[g][r] = v;
        }
    if (m == 0) {
#pragma unroll
        for (int g = 0; g < 2; ++g)
#pragma unroll
            for (int r = 0; r < 8; ++r)
                outSlice[tileBase + wave * ROWS_PER_WAVE + 16 * g + 8 * hi + r] =
                    p[g][r] + bo_total;
    }
}

__global__ __launch_bounds__(256)
void nam_reduce_kernel(const float* __restrict__ part, float* __restrict__ out)
{
    const int b = blockIdx.x * blockDim.x + threadIdx.x;
    if (b < BATCH)
        out[b] = (part[b] + part[BATCH + b]) +
                 (part[2 * BATCH + b] + part[3 * BATCH + b]);
}

extern "C" void kernel_launch(void* const* d_in, const int* in_sizes, int n_in,
                              void* d_out, int out_size, void* d_ws, size_t ws_size,
                              hipStream_t stream) {
    const float* x  = (const float*)d_in[0];
    const float* W1 = (const float*)d_in[1];
    const float* b1 = (const float*)d_in[2];
    const float* Wh = (const float*)d_in[3];
    const float* bh = (const float*)d_in[4];
    const float* Wo = (const float*)d_in[5];
    const float* bo = (const float*)d_in[6];
    float* out = (float*)d_out;

    dim3 block(THREADS);
    const size_t partBytes = (size_t)NSLICES * BATCH * sizeof(float);

    if (ws_size >= partBytes) {
        // 4 feature slices -> 512 blocks; deterministic two-stage reduction.
        float* part = (float*)d_ws;
        dim3 grid(BATCH / ROWS_PER_BLOCK, NSLICES);
        hipLaunchKernelGGL(nam_wmma_kernel, grid, block, 0, stream,
                           x, W1, b1, Wh, bh, Wo, bo, part, F_PER_SLICE);
        hipLaunchKernelGGL(nam_reduce_kernel, dim3(BATCH / 256), dim3(256), 0, stream,
                           part, out);
    } else {
        dim3 grid(BATCH / ROWS_PER_BLOCK, 1);
        hipLaunchKernelGGL(nam_wmma_kernel, grid, block, 0, stream,
                           x, W1, b1, Wh, bh, Wo, bo, out, F);
    }
}